// GraphAttentionLayer_85856396247182
// MI455X (gfx1250) — compile-verified
//
#include <hip/hip_runtime.h>
#include <hip/hip_bf16.h>
#include <math.h>

#define NN 50000
#define EE 800000
#define HH 4
#define CC 32
#define DINF 128
#define EDF 16
#define NEG_SLOPE 0.2f
#define LN_EPS 1e-5f

typedef __attribute__((ext_vector_type(2))) float v2f;
typedef __attribute__((ext_vector_type(8))) float v8f;

// ---- monotone float<->uint encoding for atomicMax on floats ----
__device__ __forceinline__ unsigned fenc(float f) {
    unsigned u = __float_as_uint(f);
    return (u & 0x80000000u) ? ~u : (u | 0x80000000u);
}
__device__ __forceinline__ float fdec(unsigned e) {
    unsigned u = (e & 0x80000000u) ? (e & 0x7FFFFFFFu) : ~e;
    return __uint_as_float(u);
}

// ---- zero two regions (scratch accumulators + d_out) ----
__global__ void k_zero2(float* __restrict__ p1, long long n1,
                        float* __restrict__ p2, long long n2) {
    long long i = (long long)blockIdx.x * blockDim.x + threadIdx.x;
    if (i < n1) p1[i] = 0.0f;
    if (i < n2) p2[i] = 0.0f;
}

// ---- M[d][h] = sum_c W_edge[d, h*C+c] * att_edge[h,c]  (kills the big edge GEMM) ----
__global__ void k_prep_M(const float* __restrict__ W_edge,
                         const float* __restrict__ att_edge,
                         float* __restrict__ M) {
    int t = threadIdx.x;
    if (t >= EDF * HH) return;
    int d = t >> 2, h = t & 3;
    float s = 0.0f;
    #pragma unroll
    for (int c = 0; c < CC; ++c)
        s += W_edge[d * (HH * CC) + h * CC + c] * att_edge[h * CC + c];
    M[d * HH + h] = s;
}

// ---- edge processor: ea = relu(edge_attr @ W_ep + b_ep), fp32 WMMA 16x16x4, K=16 ----
__global__ void k_edge_gemm(const float* __restrict__ EA, const float* __restrict__ W,
                            const float* __restrict__ bvec, float* __restrict__ ea) {
    const int lane = threadIdx.x & 31;
    const int wave = threadIdx.x >> 5;
    const int tile = blockIdx.x * 8 + wave;        // wave-uniform
    const long long m0 = (long long)tile * 16;
    if (m0 >= EE) return;                          // uniform per wave: EXEC all-ones inside
    const int lm = lane & 15;
    const int kh = (lane >> 4) << 1;               // 0 or 2
    v8f acc = {};
    const float* arow = EA + (m0 + lm) * EDF;
    #pragma unroll
    for (int k0 = 0; k0 < EDF; k0 += 4) {
        v2f a, b;
        a.x = arow[k0 + kh + 0];
        a.y = arow[k0 + kh + 1];
        b.x = W[(k0 + kh + 0) * EDF + lm];
        b.y = W[(k0 + kh + 1) * EDF + lm];
        acc = __builtin_amdgcn_wmma_f32_16x16x4_f32(false, a, false, b,
                                                    (short)0, acc, false, false);
    }
    float bb = bvec[lm];
    float* orow = ea + (m0 + (lane >> 4) * 8) * EDF + lm;
    #pragma unroll
    for (int v = 0; v < 8; ++v) {
        float t = acc[v] + bb;
        orow[v * EDF] = t > 0.0f ? t : 0.0f;
    }
}

// ---- degree + self-loop edge_attr accumulation ----
__global__ void k_scatter_loop(const long long* __restrict__ ei,
                               const float* __restrict__ ea,
                               float* __restrict__ deg, float* __restrict__ lat) {
    long long i = (long long)blockIdx.x * blockDim.x + threadIdx.x;
    if (i >= (long long)EE * EDF) return;
    int d = (int)(i & 15);
    long long e = i >> 4;
    int t = (int)ei[EE + e];
    atomicAdd(&lat[(long long)t * EDF + d], ea[e * EDF + d]);
    if (d == 0) atomicAdd(&deg[t], 1.0f);
}

// ---- h = x @ W_lin : fp32 WMMA 16x16x4, K=128. block = 16-row stripe, 8 waves = 8 col tiles ----
__global__ void k_node_gemm(const float* __restrict__ X, const float* __restrict__ W,
                            float* __restrict__ Hm) {
    const int lane = threadIdx.x & 31;
    const int wave = threadIdx.x >> 5;
    const long long m0 = (long long)blockIdx.x * 16;   // 3125 blocks, exact
    const int n0 = wave * 16;                          // 8 column tiles
    const int lm = lane & 15;
    const int kh = (lane >> 4) << 1;
    v8f acc = {};
    const float* arow = X + (m0 + lm) * DINF;
    #pragma unroll 4
    for (int k0 = 0; k0 < DINF; k0 += 4) {
        v2f a, b;
        a.x = arow[k0 + kh + 0];
        a.y = arow[k0 + kh + 1];
        b.x = W[(k0 + kh + 0) * DINF + n0 + lm];
        b.y = W[(k0 + kh + 1) * DINF + n0 + lm];
        acc = __builtin_amdgcn_wmma_f32_16x16x4_f32(false, a, false, b,
                                                    (short)0, acc, false, false);
    }
    float* orow = Hm + (m0 + (lane >> 4) * 8) * DINF + n0 + lm;
    #pragma unroll
    for (int v = 0; v < 8; ++v) orow[v * DINF] = acc[v];
}

// ---- per-(node,head): a_src, a_dst dots + self-loop a_e ----
__global__ void k_attn_node(const float* __restrict__ Hm, const float* __restrict__ att_src,
                            const float* __restrict__ att_dst, const float* __restrict__ lat,
                            const float* __restrict__ deg, const float* __restrict__ M,
                            float* __restrict__ a_src, float* __restrict__ a_dst,
                            float* __restrict__ a_el) {
    int i = blockIdx.x * blockDim.x + threadIdx.x;
    if (i >= NN * HH) return;
    int n = i >> 2, h = i & 3;
    const float* hr = Hm + (long long)n * DINF + h * CC;
    float s1 = 0.0f, s2 = 0.0f;
    #pragma unroll
    for (int c = 0; c < CC; ++c) {
        s1 += hr[c] * att_src[h * CC + c];
        s2 += hr[c] * att_dst[h * CC + c];
    }
    a_src[i] = s1;
    a_dst[i] = s2;
    float dg = deg[n];
    float inv = 1.0f / (dg > 1.0f ? dg : 1.0f);
    float s3 = 0.0f;
    #pragma unroll
    for (int d = 0; d < EDF; ++d) s3 += lat[(long long)n * EDF + d] * inv * M[d * HH + h];
    a_el[i] = s3;
}

// ---- alpha + LeakyReLU + segment max (encoded atomicMax) over E+N edges ----
__global__ void k_alpha_max(const long long* __restrict__ ei, const float* __restrict__ ea,
                            const float* __restrict__ M, const float* __restrict__ a_src,
                            const float* __restrict__ a_dst, const float* __restrict__ a_el,
                            float* __restrict__ alpha, unsigned* __restrict__ menc) {
    long long i = (long long)blockIdx.x * blockDim.x + threadIdx.x;
    if (i >= (long long)(EE + NN) * HH) return;
    int h = (int)(i & 3);
    long long e = i >> 2;
    int s, t;
    float ae;
    if (e < EE) {
        s = (int)ei[e];
        t = (int)ei[EE + e];
        const float* er = ea + e * EDF;
        float acc = 0.0f;
        #pragma unroll
        for (int d = 0; d < EDF; ++d) acc += er[d] * M[d * HH + h];
        ae = acc;
    } else {
        int n = (int)(e - EE);
        s = n; t = n;
        ae = a_el[n * HH + h];
    }
    float al = a_src[s * HH + h] + a_dst[t * HH + h] + ae;
    al = al >= 0.0f ? al : NEG_SLOPE * al;
    alpha[i] = al;
    atomicMax(&menc[t * HH + h], fenc(al));
}

// ---- exp(alpha - max) + segment sum ----
__global__ void k_exp_sum(const long long* __restrict__ ei, float* __restrict__ alpha,
                          const unsigned* __restrict__ menc, float* __restrict__ denom) {
    long long i = (long long)blockIdx.x * blockDim.x + threadIdx.x;
    if (i >= (long long)(EE + NN) * HH) return;
    int h = (int)(i & 3);
    long long e = i >> 2;
    int t = (e < EE) ? (int)ei[EE + e] : (int)(e - EE);
    float m = fdec(menc[t * HH + h]);
    float ex = __expf(alpha[i] - m);
    alpha[i] = ex;
    atomicAdd(&denom[t * HH + h], ex);
}

// ---- weighted message scatter: lane per (edge, channel); wave shares (edge, head) ----
__global__ void k_agg(const long long* __restrict__ ei, const float* __restrict__ Hm,
                      const float* __restrict__ alpha, const float* __restrict__ denom,
                      float* __restrict__ out) {
    long long i = (long long)blockIdx.x * blockDim.x + threadIdx.x;
    if (i >= (long long)(EE + NN) * DINF) return;
    int c = (int)(i & 127);
    long long e = i >> 7;
    int h = c >> 5;
    int s, t;
    if (e < EE) { s = (int)ei[e]; t = (int)ei[EE + e]; }
    else        { s = (int)(e - EE); t = s; }
    float w = alpha[e * HH + h] / denom[t * HH + h];
    atomicAdd(&out[(long long)t * DINF + c], Hm[(long long)s * DINF + c] * w);
}

// ---- residual + LayerNorm, one wave32 per row, shfl reduction ----
__global__ void k_ln(const float* __restrict__ X, const float* __restrict__ bias,
                     const float* __restrict__ g, const float* __restrict__ b,
                     float* __restrict__ out) {
    int wave = threadIdx.x >> 5, lane = threadIdx.x & 31;
    int n = blockIdx.x * (blockDim.x >> 5) + wave;
    if (n >= NN) return;
    float v[4];
    float sum = 0.0f, sq = 0.0f;
    #pragma unroll
    for (int j = 0; j < 4; ++j) {
        int c = lane + j * 32;
        float t = out[(long long)n * DINF + c] + bias[c] + X[(long long)n * DINF + c];
        v[j] = t; sum += t; sq += t * t;
    }
    #pragma unroll
    for (int m = 16; m > 0; m >>= 1) {
        sum += __shfl_xor(sum, m, 32);
        sq  += __shfl_xor(sq,  m, 32);
    }
    float mu  = sum * (1.0f / 128.0f);
    float var = sq * (1.0f / 128.0f) - mu * mu;
    float inv = rsqrtf(var + LN_EPS);
    #pragma unroll
    for (int j = 0; j < 4; ++j) {
        int c = lane + j * 32;
        out[(long long)n * DINF + c] = (v[j] - mu) * inv * g[c] + b[c];
    }
}

extern "C" void kernel_launch(void* const* d_in, const int* in_sizes, int n_in,
                              void* d_out, int out_size, void* d_ws, size_t ws_size,
                              hipStream_t stream) {
    const float*     x        = (const float*)d_in[0];
    const long long* ei       = (const long long*)d_in[1];   // int64 in reference
    const float*     eattr    = (const float*)d_in[2];
    const float*     W_ep     = (const float*)d_in[3];
    const float*     b_ep     = (const float*)d_in[4];
    const float*     W_lin    = (const float*)d_in[5];
    const float*     W_edge   = (const float*)d_in[6];
    const float*     att_src  = (const float*)d_in[7];
    const float*     att_dst  = (const float*)d_in[8];
    const float*     att_edge = (const float*)d_in[9];
    const float*     bias     = (const float*)d_in[10];
    const float*     ln_g     = (const float*)d_in[11];
    const float*     ln_b     = (const float*)d_in[12];
    float* out = (float*)d_out;

    // ---- workspace carve-up (floats, 256B-aligned chunks) ----
    float* ws = (float*)d_ws;
    size_t off = 0;
    auto alloc = [&](size_t n) { float* p = ws + off; off += (n + 63) & ~(size_t)63; return p; };
    float* ea    = alloc((size_t)EE * EDF);          // processed edge features
    float* hm    = alloc((size_t)NN * DINF);         // h = x @ W_lin
    float* a_src = alloc((size_t)NN * HH);
    float* a_dst = alloc((size_t)NN * HH);
    float* a_el  = alloc((size_t)NN * HH);
    float* alpha = alloc((size_t)(EE + NN) * HH);    // logits -> exp weights (in place)
    float* M     = alloc((size_t)EDF * HH);          // pre-contracted W_edge * att_edge
    float* zbase = ws + off;                         // contiguous zero-init region:
    float* deg   = alloc(NN);
    float* lat   = alloc((size_t)NN * EDF);          // self-loop attr accumulator
    float* denom = alloc((size_t)NN * HH);
    unsigned* menc = (unsigned*)alloc((size_t)NN * HH);  // encoded segment max (0 == -inf floor)
    long long nz = (long long)((ws + off) - zbase);

    const long long nout = (long long)NN * DINF;
    const long long nzmax = nz > nout ? nz : nout;
    k_zero2<<<(unsigned)((nzmax + 255) / 256), 256, 0, stream>>>(zbase, nz, out, nout);

    k_prep_M<<<1, 64, 0, stream>>>(W_edge, att_edge, M);

    k_edge_gemm<<<(unsigned)(EE / 16 / 8), 256, 0, stream>>>(eattr, W_ep, b_ep, ea);

    k_scatter_loop<<<(unsigned)(((long long)EE * EDF + 255) / 256), 256, 0, stream>>>(
        ei, ea, deg, lat);

    k_node_gemm<<<(unsigned)(NN / 16), 256, 0, stream>>>(x, W_lin, hm);

    k_attn_node<<<(unsigned)((NN * HH + 255) / 256), 256, 0, stream>>>(
        hm, att_src, att_dst, lat, deg, M, a_src, a_dst, a_el);

    const long long nah = (long long)(EE + NN) * HH;
    k_alpha_max<<<(unsigned)((nah + 255) / 256), 256, 0, stream>>>(
        ei, ea, M, a_src, a_dst, a_el, alpha, menc);

    k_exp_sum<<<(unsigned)((nah + 255) / 256), 256, 0, stream>>>(ei, alpha, menc, denom);

    const long long nag = (long long)(EE + NN) * DINF;
    k_agg<<<(unsigned)((nag + 255) / 256), 256, 0, stream>>>(ei, hm, alpha, denom, out);

    k_ln<<<(unsigned)((NN + 7) / 8), 256, 0, stream>>>(x, bias, ln_g, ln_b, out);
}